// Laplacianloss_77506979823874
// MI455X (gfx1250) — compile-verified
//
#include <hip/hip_runtime.h>
#include <hip/hip_bf16.h>
#include <math.h>

typedef __attribute__((ext_vector_type(2))) float v2f;
typedef __attribute__((ext_vector_type(8))) float v8f;
typedef __attribute__((ext_vector_type(4))) int   v4i;

// ---------------------------------------------------------------------------
// Wave32 reduction via V_WMMA_F32_16X16X4_F32 against an all-ones B matrix.
// A(16x4 f32) = 2 VGPRs x 32 lanes: lane m (0-15) supplies A[m][0..1],
// lane m+16 supplies A[m][2..3]. With B = ones, D[m][*] = rowsum(m).
// Summing the 8 D VGPRs leaves:
//   lanes 0-15 : sum of lanes {0..7, 16..23}   (S_lo)
//   lanes 16-31: sum of lanes {8..15, 24..31}  (S_hi)
// S_lo + S_hi = full wave sum. Lane 0 and lane 16 each deposit their half.
// ---------------------------------------------------------------------------
__device__ __forceinline__ float wave_wmma_halfsum(float s) {
  v2f a;
  a.x = s;
  a.y = 0.0f;
  v2f ones;
  ones.x = 1.0f;
  ones.y = 1.0f;
  v8f c = {};
  v8f dm = __builtin_amdgcn_wmma_f32_16x16x4_f32(
      /*neg_a=*/false, a, /*neg_b=*/false, ones,
      /*c_mod=*/(short)0, c, /*reuse_a=*/false, /*reuse_b=*/false);
  return dm[0] + dm[1] + dm[2] + dm[3] + dm[4] + dm[5] + dm[6] + dm[7];
}

// ---------------------------------------------------------------------------
// Kernel 1: zero the 3N-float neighbor-sum accumulator.
// Must run every launch: harness does not re-zero workspace between replays.
// ---------------------------------------------------------------------------
__global__ void lap_zero_kernel(float* __restrict__ p, long n) {
  long i = (long)blockIdx.x * blockDim.x + threadIdx.x;
  long stride = (long)gridDim.x * blockDim.x;
  for (; i < n; i += stride) p[i] = 0.0f;
}

// ---------------------------------------------------------------------------
// Kernel 2: edge scatter. For edge e: acc[dst] += v[src]  (segment_sum).
// Edge stream is coalesced (int4 = 2 edges/lane) and read-once -> nontemporal.
// v[src] gathers + acc atomics hit L2-resident arrays (12MB each vs 192MB L2).
// Atomics are spread over 3M addresses: no serialization hotspot.
// ---------------------------------------------------------------------------
__device__ __forceinline__ void scatter_one(int dst, int src,
                                            const float* __restrict__ v,
                                            float* __restrict__ acc) {
  float x = v[3 * (long)src + 0];
  float y = v[3 * (long)src + 1];
  float z = v[3 * (long)src + 2];
  float* a = acc + 3 * (long)dst;
  __hip_atomic_fetch_add(a + 0, x, __ATOMIC_RELAXED, __HIP_MEMORY_SCOPE_AGENT);
  __hip_atomic_fetch_add(a + 1, y, __ATOMIC_RELAXED, __HIP_MEMORY_SCOPE_AGENT);
  __hip_atomic_fetch_add(a + 2, z, __ATOMIC_RELAXED, __HIP_MEMORY_SCOPE_AGENT);
}

__global__ void lap_scatter_kernel(const int* __restrict__ edges,
                                   const float* __restrict__ v,
                                   float* __restrict__ acc,
                                   int nedges) {
  int t = blockIdx.x * blockDim.x + threadIdx.x;
  int e0 = t * 2;
  if (e0 >= nedges) return;
  if (e0 + 1 < nedges) {
    const v4i ee = __builtin_nontemporal_load(((const v4i*)edges) + t);
    scatter_one(ee.x, ee.y, v, acc);
    scatter_one(ee.z, ee.w, v, acc);
  } else {
    scatter_one(edges[2 * e0], edges[2 * e0 + 1], v, acc);
  }
}

// ---------------------------------------------------------------------------
// Kernel 3: lap = deg*v - acc; block-reduce ||lap||^2 and write one partial
// per block (plain store -> no same-address atomic hotspot, deterministic).
// ---------------------------------------------------------------------------
__global__ void lap_norm_kernel(const float* __restrict__ v,
                                const float* __restrict__ deg,
                                const float* __restrict__ acc,
                                float* __restrict__ partials, int n) {
  __shared__ float lds[16];  // 8 waves x 2 half-sums
  int i = blockIdx.x * blockDim.x + threadIdx.x;
  float s = 0.0f;
  if (i < n) {
    float d = deg[i];
    long b = 3 * (long)i;
    float lx = fmaf(d, v[b + 0], -acc[b + 0]);
    float ly = fmaf(d, v[b + 1], -acc[b + 1]);
    float lz = fmaf(d, v[b + 2], -acc[b + 2]);
    s = lx * lx + ly * ly + lz * lz;
  }

  float w = wave_wmma_halfsum(s);            // EXEC full here (reconverged)
  if ((threadIdx.x & 15) == 0)               // lanes 0 and 16 of each wave
    lds[threadIdx.x >> 4] = w;
  __syncthreads();

  if (threadIdx.x == 0) {
    float t = 0.0f;
#pragma unroll
    for (int k = 0; k < 16; ++k) t += lds[k];
    partials[blockIdx.x] = t;
  }
}

// ---------------------------------------------------------------------------
// Kernel 4: one block reduces all block partials (deterministic order within
// each lane + exact f32 WMMA combine) and writes sqrt to d_out.
// ---------------------------------------------------------------------------
__global__ void lap_finalize_kernel(const float* __restrict__ partials,
                                    int nparts, float* __restrict__ out) {
  __shared__ float lds[16];
  float s = 0.0f;
  for (int i = threadIdx.x; i < nparts; i += blockDim.x) s += partials[i];

  float w = wave_wmma_halfsum(s);
  if ((threadIdx.x & 15) == 0) lds[threadIdx.x >> 4] = w;
  __syncthreads();

  if (threadIdx.x == 0) {
    float t = 0.0f;
#pragma unroll
    for (int k = 0; k < 16; ++k) t += lds[k];
    out[0] = sqrtf(t);
  }
}

extern "C" void kernel_launch(void* const* d_in, const int* in_sizes, int n_in,
                              void* d_out, int out_size, void* d_ws, size_t ws_size,
                              hipStream_t stream) {
  const float* vertices = (const float*)d_in[0];  // (1, N, 3) f32
  const float* deg      = (const float*)d_in[1];  // (N,) f32
  const int*   edges    = (const int*)d_in[2];    // (E, 2) i32

  const int n      = in_sizes[1];       // N vertices
  const int nedges = in_sizes[2] / 2;   // E directed edges

  const int BS = 256;
  const int nblocks = (n + BS - 1) / BS;

  float* acc      = (float*)d_ws;               // 3N floats (neighbor sums)
  float* partials = acc + 3 * (size_t)n;        // nblocks floats

  // 1) zero the accumulator
  long zn = 3L * n;
  int zb = (int)((zn + BS - 1) / BS);
  if (zb > 4096) zb = 4096;
  lap_zero_kernel<<<zb, BS, 0, stream>>>(acc, zn);

  // 2) atomic segment-sum over edges (2 edges / thread)
  int pairs = (nedges + 1) / 2;
  lap_scatter_kernel<<<(pairs + BS - 1) / BS, BS, 0, stream>>>(edges, vertices,
                                                               acc, nedges);

  // 3) squared Frobenius norm -> one partial per block (WMMA wave reduction)
  lap_norm_kernel<<<nblocks, BS, 0, stream>>>(vertices, deg, acc, partials, n);

  // 4) single-block reduction of partials + sqrt -> d_out
  lap_finalize_kernel<<<1, BS, 0, stream>>>(partials, nblocks, (float*)d_out);
}